// JSDCL_60730837566262
// MI455X (gfx1250) — compile-verified
//
#include <hip/hip_runtime.h>
#include <math.h>

#define N_ROWS 512
#define D_DIM  768
#define EPSF   1e-8f
#define LAMBDF 1.0f

#define TIB 32      // 32 i-rows x 32 j-rows of pairs per block (2x2 register blocking)
#define DK  256     // d-chunk staged in LDS
#define DKP 260     // padded LDS row stride (floats): 260 % 64 = 4 -> conflict-free across rows

#define L2EF      1.44269504088896340736f   // log2(e)
#define LN2F      0.69314718055994530942f
#define TWO_LN2F  1.38629436111989061884f

typedef __attribute__((ext_vector_type(2))) float v2f;
typedef __attribute__((ext_vector_type(8))) float v8f;

// Native transcendental wrappers: v_exp_f32 computes 2^x, v_log_f32 computes log2(x).
__device__ __forceinline__ float fast_exp2(float x) {
#if __has_builtin(__builtin_amdgcn_exp2f)
  return __builtin_amdgcn_exp2f(x);
#else
  return exp2f(x);
#endif
}
__device__ __forceinline__ float fast_log2(float x) {
#if __has_builtin(__builtin_amdgcn_logf)
  return __builtin_amdgcn_logf(x);
#else
  return log2f(x);
#endif
}

// ---------------- wave / block reductions (wave32, fixed order, deterministic) ----------

__device__ __forceinline__ float wave_reduce_sum(float v) {
#pragma unroll
  for (int off = 16; off > 0; off >>= 1) v += __shfl_xor(v, off, 32);
  return v;
}
__device__ __forceinline__ float wave_reduce_max(float v) {
#pragma unroll
  for (int off = 16; off > 0; off >>= 1) v = fmaxf(v, __shfl_xor(v, off, 32));
  return v;
}
__device__ __forceinline__ float block_reduce_sum(float v, float* sm8) {
  const int lane = threadIdx.x & 31, wid = threadIdx.x >> 5;
  v = wave_reduce_sum(v);
  if (lane == 0) sm8[wid] = v;
  __syncthreads();
  float t = (lane < 8) ? sm8[lane] : 0.0f;
  t = wave_reduce_sum(t);          // all waves reduce the 8 partials -> broadcast
  __syncthreads();
  return t;
}
__device__ __forceinline__ float block_reduce_max(float v, float* sm8) {
  const int lane = threadIdx.x & 31, wid = threadIdx.x >> 5;
  v = wave_reduce_max(v);
  if (lane == 0) sm8[wid] = v;
  __syncthreads();
  float t = (lane < 8) ? sm8[lane] : -INFINITY;
  t = wave_reduce_max(t);
  __syncthreads();
  return t;
}

// ---------------- kernel 1: softmax stats (max, 1/denominator) + row entropy ------------
// blockIdx.x in [0,1024): rows 0..511 -> z1, 512..1023 -> z2.
__global__ __launch_bounds__(256) void jsd_softmax_stats(
    const float* __restrict__ z1, const float* __restrict__ z2,
    float* __restrict__ m_all, float* __restrict__ r_all, float* __restrict__ H_all) {
  __shared__ float sm8[8];
  const int row = blockIdx.x;
  const float* z = (row < N_ROWS) ? (z1 + (size_t)row * D_DIM)
                                  : (z2 + (size_t)(row - N_ROWS) * D_DIM);
  const int tid = threadIdx.x;
  const float v0 = z[tid], v1 = z[tid + 256], v2 = z[tid + 512];

  const float mx = block_reduce_max(fmaxf(v0, fmaxf(v1, v2)), sm8);
  // exp(x) = exp2(x * log2(e)); identical expression is reused in kernel 2.
  const float e0 = fast_exp2((v0 - mx) * L2EF);
  const float e1 = fast_exp2((v1 - mx) * L2EF);
  const float e2 = fast_exp2((v2 - mx) * L2EF);
  const float den = block_reduce_sum(e0 + e1 + e2, sm8);
  const float rinv = 1.0f / den;
  const float p0 = e0 * rinv, p1 = e1 * rinv, p2 = e2 * rinv;
  // H = sum p*ln(p+eps) accumulated in log2 domain, scaled by ln2 once.
  float h = p0 * fast_log2(p0 + EPSF) + p1 * fast_log2(p1 + EPSF) + p2 * fast_log2(p2 + EPSF);
  h = block_reduce_sum(h, sm8);
  if (tid == 0) { m_all[row] = mx; r_all[row] = rinv; H_all[row] = h * LN2F; }
}

// ---------------- kernel 2: pairwise cross term --------------------------------------
// C[i,j] = sum_d (p1+p2) ln(0.5(p1+p2)+eps) = 2*ln2 * sum_d s'*log2(s'+eps), s' = 0.5(p1+p2).
// LDS holds 0.5*p tiles. Grid 16x16; each thread owns a 2x2 pair block {li,li+16}x{lj,lj+16}.
__global__ __launch_bounds__(256) void jsd_cross(
    const float* __restrict__ z1, const float* __restrict__ z2,
    const float* __restrict__ m_all, const float* __restrict__ r_all,
    float* __restrict__ blockS, float* __restrict__ blockD) {
  __shared__ float p1L[TIB * DKP];
  __shared__ float p2L[TIB * DKP];
  __shared__ float red[8];
  __shared__ float redD[TIB];

  const int tid = threadIdx.x;
  const int bj = blockIdx.x, bi = blockIdx.y;
  const int i0 = bi * TIB, j0 = bj * TIB;
  const int li = tid >> 4, lj = tid & 15;

  float acc00 = 0.0f, acc01 = 0.0f, acc10 = 0.0f, acc11 = 0.0f;

  for (int dc = 0; dc < D_DIM; dc += DK) {
    // Stage half-softmax tiles (0.5*p) into LDS; per-row stats are uniform -> SMEM loads.
#pragma unroll
    for (int r = 0; r < TIB; ++r) {
      const float a = fast_exp2((z1[(size_t)(i0 + r) * D_DIM + dc + tid] - m_all[i0 + r]) * L2EF)
                      * (0.5f * r_all[i0 + r]);
      p1L[r * DKP + tid] = a;
      const float b = fast_exp2((z2[(size_t)(j0 + r) * D_DIM + dc + tid] - m_all[N_ROWS + j0 + r]) * L2EF)
                      * (0.5f * r_all[N_ROWS + j0 + r]);
      p2L[r * DKP + tid] = b;
    }
    __syncthreads();

    if (dc + DK < D_DIM) {
      __builtin_prefetch(z1 + (size_t)(i0 + li) * D_DIM + dc + DK + lj * 16, 0, 1);
      __builtin_prefetch(z2 + (size_t)(j0 + li) * D_DIM + dc + DK + lj * 16, 0, 1);
      __builtin_prefetch(z1 + (size_t)(i0 + 16 + li) * D_DIM + dc + DK + lj * 16, 0, 1);
      __builtin_prefetch(z2 + (size_t)(j0 + 16 + li) * D_DIM + dc + DK + lj * 16, 0, 1);
    }

    const float* pa0 = &p1L[li * DKP];
    const float* pa1 = &p1L[(li + 16) * DKP];
    const float* pb0 = &p2L[lj * DKP];
    const float* pb1 = &p2L[(lj + 16) * DKP];
#pragma unroll 2
    for (int d = 0; d < DK; d += 4) {
      const float4 a0 = *(const float4*)(pa0 + d);   // ds_load_b128, conflict-free (stride 260)
      const float4 a1 = *(const float4*)(pa1 + d);
      const float4 b0 = *(const float4*)(pb0 + d);
      const float4 b1 = *(const float4*)(pb1 + d);
      float s;
      // per element: v_add + v_log_f32 (TRANS) + v_fmac -> sits on the TRANS ceiling
      s = a0.x + b0.x; acc00 = fmaf(s, fast_log2(s + EPSF), acc00);
      s = a0.x + b1.x; acc01 = fmaf(s, fast_log2(s + EPSF), acc01);
      s = a1.x + b0.x; acc10 = fmaf(s, fast_log2(s + EPSF), acc10);
      s = a1.x + b1.x; acc11 = fmaf(s, fast_log2(s + EPSF), acc11);
      s = a0.y + b0.y; acc00 = fmaf(s, fast_log2(s + EPSF), acc00);
      s = a0.y + b1.y; acc01 = fmaf(s, fast_log2(s + EPSF), acc01);
      s = a1.y + b0.y; acc10 = fmaf(s, fast_log2(s + EPSF), acc10);
      s = a1.y + b1.y; acc11 = fmaf(s, fast_log2(s + EPSF), acc11);
      s = a0.z + b0.z; acc00 = fmaf(s, fast_log2(s + EPSF), acc00);
      s = a0.z + b1.z; acc01 = fmaf(s, fast_log2(s + EPSF), acc01);
      s = a1.z + b0.z; acc10 = fmaf(s, fast_log2(s + EPSF), acc10);
      s = a1.z + b1.z; acc11 = fmaf(s, fast_log2(s + EPSF), acc11);
      s = a0.w + b0.w; acc00 = fmaf(s, fast_log2(s + EPSF), acc00);
      s = a0.w + b1.w; acc01 = fmaf(s, fast_log2(s + EPSF), acc01);
      s = a1.w + b0.w; acc10 = fmaf(s, fast_log2(s + EPSF), acc10);
      s = a1.w + b1.w; acc11 = fmaf(s, fast_log2(s + EPSF), acc11);
    }
    __syncthreads();
  }

  const float accT = (acc00 + acc01) + (acc10 + acc11);

  // ---- per-wave reduction on the matrix pipe: D = A(partials) x B(ones) + 0 ----
  const int lane = tid & 31, wid = tid >> 5;
#if __has_builtin(__builtin_amdgcn_wmma_f32_16x16x4_f32)
  v2f A; A[0] = accT; A[1] = 0.0f;     // 32 partials in the wave's 16x4 A tile (rest zero)
  v2f B; B[0] = 1.0f; B[1] = 1.0f;     // all-ones B => D[m,n] = sum_k A[m,k]
  v8f C = {};
  v8f Dv = __builtin_amdgcn_wmma_f32_16x16x4_f32(false, A, false, B, (short)0, C, false, false);
  float sum8 = ((Dv[0] + Dv[1]) + (Dv[2] + Dv[3])) + ((Dv[4] + Dv[5]) + (Dv[6] + Dv[7]));
  const float wtot = sum8 + __shfl_xor(sum8, 16, 32);   // combine M=0..7 / M=8..15 halves
#else
  const float wtot = wave_reduce_sum(accT);
#endif
  if (lane == 0) red[wid] = wtot;

  const bool diagBlock = (bi == bj);
  if (diagBlock && (li == lj)) { redD[li] = acc00; redD[li + 16] = acc11; }
  __syncthreads();

  if (tid == 0) {
    float s = 0.0f;
#pragma unroll
    for (int w = 0; w < 8; ++w) s += red[w];
    blockS[bi * 16 + bj] = s * TWO_LN2F;
    if (diagBlock) {
      float ds = 0.0f;
#pragma unroll
      for (int k = 0; k < TIB; ++k) ds += redD[k];
      blockD[bi] = ds * TWO_LN2F;
    }
  }
}

// ---------------- kernel 3: deterministic final combine -> scalar -----------------------
__global__ __launch_bounds__(256) void jsd_final(
    const float* __restrict__ H_all, const float* __restrict__ bS,
    const float* __restrict__ bD, float* __restrict__ out) {
  __shared__ float sm8[8];
  const int tid = threadIdx.x;
  float h = 0.0f;
  for (int k = tid; k < 1024; k += 256) h += H_all[k];
  float s = bS[tid];                       // exactly 256 block partials
  float d = (tid < 16) ? bD[tid] : 0.0f;   // 16 diagonal-block partials
  h = block_reduce_sum(h, sm8);   // SH = sum H1 + sum H2
  s = block_reduce_sum(s, sm8);   // SC = sum_{ij} C
  d = block_reduce_sum(d, sm8);   // DC = sum_i C[i,i]
  if (tid == 0) {
    const float n = (float)N_ROWS;
    const float Dsum = 0.5f * (h - d);            // sum_i jsd[i,i]
    const float Sall = 0.5f * (n * h - s);        // sum_{ij} jsd[i,j]
    const float pos = Dsum / n;
    const float neg = -(Sall - Dsum) / (n * n - n);
    out[0] = pos + LAMBDF * neg;
  }
}

extern "C" void kernel_launch(void* const* d_in, const int* in_sizes, int n_in,
                              void* d_out, int out_size, void* d_ws, size_t ws_size,
                              hipStream_t stream) {
  (void)in_sizes; (void)n_in; (void)out_size; (void)ws_size;
  const float* z1 = (const float*)d_in[0];
  const float* z2 = (const float*)d_in[1];
  float* ws = (float*)d_ws;
  float* m_all  = ws;            // 1024 floats (rows of z1 then z2)
  float* r_all  = ws + 1024;     // 1024
  float* H_all  = ws + 2048;     // 1024
  float* blockS = ws + 3072;     // 256 (per-block cross partials)
  float* blockD = ws + 3328;     // 16  (diagonal-block cross partials)

  jsd_softmax_stats<<<1024, 256, 0, stream>>>(z1, z2, m_all, r_all, H_all);
  dim3 grid(16, 16);
  jsd_cross<<<grid, 256, 0, stream>>>(z1, z2, m_all, r_all, blockS, blockD);
  jsd_final<<<1, 256, 0, stream>>>(H_all, blockS, blockD, (float*)d_out);
}